// YOLOv3_9070970929598
// MI455X (gfx1250) — compile-verified
//
#include <hip/hip_runtime.h>
#include <math.h>

typedef __attribute__((ext_vector_type(2))) float v2f;
typedef __attribute__((ext_vector_type(8))) float v8f;

#define NA 3
#define HH 76
#define WW 76
#define NCLS 80
#define STRIDE_F 8.0f
#define IGNORE_THRE 0.6f
#define MAXG 64

__constant__ float c_anchors[9][2] = {
  {10.f,13.f},{16.f,30.f},{33.f,23.f},{30.f,61.f},{62.f,45.f},
  {59.f,119.f},{116.f,90.f},{156.f,198.f},{373.f,326.f}
};

__device__ __forceinline__ float bce(float x, float z) {
  // max(x,0) - x*z + log1p(exp(-|x|))
  return fmaxf(x, 0.f) - x * z + log1pf(expf(-fabsf(x)));
}

__device__ __forceinline__ float sigmoidf_(float x) {
  return 1.f / (1.f + expf(-x));
}

__global__ __launch_bounds__(256) void yolo_loss_kernel(
    const float* __restrict__ t_xywh,
    const float* __restrict__ conf_logits,
    const float* __restrict__ cls_logits,
    const float* __restrict__ labels,
    const void*  __restrict__ img_size_p,
    float* __restrict__ out,
    int G, int chunks_per_img)
{
  __shared__ int   sh_valid[MAXG], sh_a[MAXG], sh_ti[MAXG], sh_tj[MAXG], sh_cls[MAXG];
  __shared__ float sh_tx[MAXG], sh_ty[MAXG], sh_tw[MAXG], sh_th[MAXG], sh_wgt[MAXG];
  __shared__ float sh_gx[MAXG], sh_gy[MAXG], sh_gw[MAXG], sh_gh[MAXG];
  __shared__ int   sh_has_valid;
  __shared__ float sh_wsum[8];

  const int tid   = threadIdx.x;
  const int b     = blockIdx.x / chunks_per_img;
  const int chunk = blockIdx.x % chunks_per_img;

  if (tid == 0) sh_has_valid = 0;
  __syncthreads();

  // img_size scalar: tolerate int32 or float32 encoding
  float img;
  {
    int bits = *(const int*)img_size_p;
    if (bits > 0 && bits < (1 << 20)) {
      img = (float)bits;
    } else {
      float f; __builtin_memcpy(&f, &bits, 4); img = f;
    }
  }

  // ---- Phase 1: per-label target assignment into LDS ----
  if (tid < G) {
    const float* lab = labels + ((size_t)b * G + tid) * 5;
    float cls = lab[0], x = lab[1], y = lab[2], w = lab[3], h = lab[4];
    // anchor IoU at origin: inter = min(w,aw)*min(h,ah); argmax (first max wins)
    int best = 0; float biou = -1.f;
    #pragma unroll
    for (int k = 0; k < 9; ++k) {
      float aw = c_anchors[k][0], ah = c_anchors[k][1];
      float inter = fminf(w, aw) * fminf(h, ah);
      float v = inter / (w * h + aw * ah - inter);
      if (v > biou) { biou = v; best = k; }
    }
    int valid = (best < 3) ? 1 : 0;          // LEVEL == 0
    int bn = best % NA;
    float gx = x * (1.f / STRIDE_F), gy = y * (1.f / STRIDE_F);
    int ti = (int)gx, tj = (int)gy;
    float aw = c_anchors[bn][0], ah = c_anchors[bn][1];
    sh_valid[tid] = valid; sh_a[tid] = bn; sh_ti[tid] = ti; sh_tj[tid] = tj;
    sh_cls[tid] = (int)cls;
    sh_tx[tid] = gx - floorf(gx);
    sh_ty[tid] = gy - floorf(gy);
    sh_tw[tid] = logf(w / aw + 1e-8f);
    sh_th[tid] = logf(h / ah + 1e-8f);
    sh_wgt[tid] = 2.f - w * h / (img * img);
    sh_gx[tid] = x; sh_gy[tid] = y; sh_gw[tid] = w; sh_gh[tid] = h;
    if (valid) atomicOr(&sh_has_valid, 1);
  }
  __syncthreads();

  // ---- Phase 2: one thread per grid cell ----
  float s = 0.f;
  const int cells = NA * HH * WW;
  int cell = chunk * 256 + tid;
  if (b < gridDim.x && cell < cells) {
    int a   = cell / (HH * WW);
    int rem = cell - a * (HH * WW);
    int j   = rem / WW;
    int i   = rem - j * WW;
    size_t base = ((size_t)(b * NA + a) * HH + j) * WW + i;

    float4 t = *(const float4*)(t_xywh + base * 4);
    float px = (sigmoidf_(t.x) + (float)i) * STRIDE_F;
    float py = (sigmoidf_(t.y) + (float)j) * STRIDE_F;
    float pw = expf(t.z) * c_anchors[a][0];
    float ph = expf(t.w) * c_anchors[a][1];
    float p_l = px - pw * 0.5f, p_r = px + pw * 0.5f;
    float p_t = py - ph * 0.5f, p_b = py + ph * 0.5f;
    float p_area = pw * ph;

    // max IoU vs all GT boxes -> ignore mask
    float miou = 0.f;
    for (int g = 0; g < G; ++g) {
      float gw = sh_gw[g], gh = sh_gh[g];
      float gl  = sh_gx[g] - gw * 0.5f, gr = sh_gx[g] + gw * 0.5f;
      float gtp = sh_gy[g] - gh * 0.5f, gb = sh_gy[g] + gh * 0.5f;
      float iw = fmaxf(fminf(p_r, gr) - fmaxf(p_l, gl), 0.f);
      float ih = fmaxf(fminf(p_b, gb) - fmaxf(p_t, gtp), 0.f);
      float inter = iw * ih;
      miou = fmaxf(miou, inter / (p_area + gw * gh - inter));
    }
    bool ign = miou < IGNORE_THRE;

    // scatter-match: which labels target this cell (last match wins for xywh/wgt,
    // OR of one-hots for the class target, matching the 4-D scatter)
    int m = -1;
    unsigned cm0 = 0, cm1 = 0, cm2 = 0;
    for (int g = 0; g < G; ++g) {
      if (sh_valid[g] && sh_a[g] == a && sh_tj[g] == j && sh_ti[g] == i) {
        m = g;
        int c = sh_cls[g];
        if (c < 32)      cm0 |= 1u << c;
        else if (c < 64) cm1 |= 1u << (c - 32);
        else             cm2 |= 1u << (c - 64);
      }
    }
    bool pos = (m >= 0);

    float cf = conf_logits[base];
    bool cmsk = sh_has_valid ? (ign || pos) : true;
    if (cmsk) s += bce(cf, pos ? 1.f : 0.f);

    if (pos) {
      const float* cp = cls_logits + base * NCLS;
      __builtin_prefetch(cp, 0, 1);            // global_prefetch_b8
      float wg = sh_wgt[m];
      s += wg * (bce(t.x, sh_tx[m]) + bce(t.y, sh_ty[m]));
      float dw = t.z - sh_tw[m], dh = t.w - sh_th[m];
      s += 0.5f * wg * (dw * dw + dh * dh);
      #pragma unroll 4
      for (int k = 0; k < NCLS; ++k) {
        unsigned bit = (k < 32) ? (cm0 >> k)
                     : (k < 64) ? (cm1 >> (k - 32))
                                : (cm2 >> (k - 64));
        s += bce(cp[k], (float)(bit & 1u));
      }
    }
  }

  // ---- Wave reduction via V_WMMA_F32_16X16X4_F32: D = A(16x4) * ones(4x16) ----
  // Lane L<16 holds A[L][0..1] = {s,0}; lane L>=16 holds A[L-16][2..3] = {s,0}.
  // D[m][n] = s_m + s_{m+16}; summing the 8 accumulator VGPRs gives rows 0-7 on
  // lanes 0-15 and rows 8-15 on lanes 16-31 -> wave total = lane0 + lane16.
  v2f av; av.x = s;   av.y = 0.f;
  v2f bv; bv.x = 1.f; bv.y = 1.f;
  v8f cv = {0.f, 0.f, 0.f, 0.f, 0.f, 0.f, 0.f, 0.f};
  cv = __builtin_amdgcn_wmma_f32_16x16x4_f32(false, av, false, bv,
                                             (short)0, cv, false, false);
  float tsum = cv[0] + cv[1] + cv[2] + cv[3] + cv[4] + cv[5] + cv[6] + cv[7];
  float wtot = __shfl(tsum, 0, 32) + __shfl(tsum, 16, 32);

  int wid = tid >> 5;
  if ((tid & 31) == 0) sh_wsum[wid] = wtot;
  __syncthreads();
  if (tid == 0) {
    float blk = 0.f;
    #pragma unroll
    for (int wv = 0; wv < 8; ++wv) blk += sh_wsum[wv];
    atomicAdd(out, blk);
  }
}

extern "C" void kernel_launch(void* const* d_in, const int* in_sizes, int n_in,
                              void* d_out, int out_size, void* d_ws, size_t ws_size,
                              hipStream_t stream) {
  const float* t_xywh = (const float*)d_in[0];
  const float* conf   = (const float*)d_in[1];
  const float* clsl   = (const float*)d_in[2];
  const float* labels = (const float*)d_in[3];
  const void*  img    = d_in[4];
  float* out = (float*)d_out;

  int B = in_sizes[0] / (NA * HH * WW * 4);
  if (B < 1) B = 1;
  int G = in_sizes[3] / (B * 5);
  if (G > MAXG) G = MAXG;
  if (G < 1) G = 1;

  const int cells = NA * HH * WW;
  const int chunks = (cells + 255) / 256;

  hipMemsetAsync(out, 0, sizeof(float), stream);  // capture-safe zero of scalar
  yolo_loss_kernel<<<B * chunks, 256, 0, stream>>>(t_xywh, conf, clsl, labels,
                                                   img, out, G, chunks);
}